// VecLinear_14972255994161
// MI455X (gfx1250) — compile-verified
//
#include <hip/hip_runtime.h>

// ---------------------------------------------------------------------------
// CDNA5 / gfx1250 fused SE(3) vector-neuron layer.
// - Channel-mixing einsums on V_WMMA_F32_16X16X4_F32 (f32 WMMA, wave32)
// - Tile staging via Tensor Data Mover (TENSOR_LOAD_TO_LDS, TENSORcnt)
// B=16, C=128, N=4096. One workgroup = 1 batch x 16-column tile, all 128 rows.
// ---------------------------------------------------------------------------

#define EPSN 1e-12f

typedef float v2f __attribute__((ext_vector_type(2)));
typedef float v8f __attribute__((ext_vector_type(8)));
typedef unsigned int u32x4 __attribute__((ext_vector_type(4)));
typedef int i32x8 __attribute__((ext_vector_type(8)));
typedef int i32x4 __attribute__((ext_vector_type(4)));

// D = A(16x4) * B(4x16) + C   -- f32 WMMA, wave32
__device__ __forceinline__ v8f wmma4(v2f a, v2f b, v8f c) {
  return __builtin_amdgcn_wmma_f32_16x16x4_f32(
      /*neg_a=*/false, a, /*neg_b=*/false, b,
      /*c_mod=*/(short)0, c, /*reuse_a=*/false, /*reuse_b=*/false);
}

// A fragment (16x4 f32): lanes 0-15 hold rows M with K={0,1}, lanes 16-31 K={2,3}
__device__ __forceinline__ v2f load_a(const float* __restrict__ W, int ldw,
                                      int m, int kk, int duo) {
  v2f a;
  a.x = W[m * ldw + kk + 2 * duo + 0];
  a.y = W[m * ldw + kk + 2 * duo + 1];
  return a;
}

// B fragment (4x16 f32): VGPR j holds row K=j (lanes 0-15) / K=j+2 (lanes 16-31)
__device__ __forceinline__ v2f load_b(const float* Bl, int ldb,
                                      int n, int kk, int duo) {
  v2f b;
  b.x = Bl[(kk + 2 * duo + 0) * ldb + n];
  b.y = Bl[(kk + 2 * duo + 1) * ldb + n];
  return b;
}

// C/D 16x16 f32: VGPR r => row r (lanes 0-15) / row r+8 (lanes 16-31)
__device__ __forceinline__ void store_tile(float* dst, int ld, int o0, int duo,
                                           int n, v8f acc) {
#pragma unroll
  for (int r = 0; r < 8; ++r) dst[(o0 + r + 8 * duo) * ld + n] = acc[r];
}

// ---------------------------------------------------------------------------
// TDM: issue TENSOR_LOAD_TO_LDS for a (up to 3-D) f32 tile.
// D# bitfields per CDNA5 ISA ch.8 (group0 128b, group1 256b, group2/3 128b).
// dims/strides in elements (data_size = 4 bytes).
// This toolchain exposes the 6-arg builtin:
//   (u32x4 g0, i32x8 g1, i32x4 g2, i32x4 g3, i32x8 extra, i32 cpol)
// ---------------------------------------------------------------------------
__device__ __forceinline__ void tdm_load_f32_tile(
    unsigned lds_byte_off, const void* gaddr,
    unsigned tile0, unsigned tile1, unsigned tile2,
    unsigned long long tdim0, unsigned long long tdim1, unsigned tdim2,
    unsigned long long stride0, unsigned long long stride1) {
  unsigned long long ga = (unsigned long long)gaddr;
  u32x4 g0;
  g0[0] = 1u;                                        // count=1, user descriptor
  g0[1] = lds_byte_off;                              // lds_addr [63:32]
  g0[2] = (unsigned)(ga & 0xffffffffu);              // global_addr[31:0]
  g0[3] = (unsigned)((ga >> 32) & 0x01ffffffu)       // global_addr[56:32]
        | (2u << 30);                                // type = 2 ("image")
  i32x8 g1;
  g1[0] = (int)(2u << 16);                           // wg_mask=0, data_size=4B
  g1[1] = (int)((tdim0 & 0xffffu) << 16);            // tensor_dim0[15:0]
  g1[2] = (int)(((tdim0 >> 16) & 0xffffu) | ((tdim1 & 0xffffu) << 16));
  g1[3] = (int)(((tdim1 >> 16) & 0xffffu) | ((tile0 & 0xffffu) << 16));
  g1[4] = (int)((tile1 & 0xffffu) | ((tile2 & 0xffffu) << 16));
  g1[5] = (int)(stride0 & 0xffffffffu);              // tensor_dim0_stride lo32
  g1[6] = (int)(((stride0 >> 32) & 0xffffu) |        // tensor_dim0_stride hi16
                ((stride1 & 0xffffu) << 16));        // tensor_dim1_stride lo16
  g1[7] = (int)((stride1 >> 16) & 0xffffffffu);      // tensor_dim1_stride hi32
  i32x4 g2;
  g2[0] = (int)tdim2;                                // tensor_dim2
  g2[1] = 0;                                         // tensor_dim3
  g2[2] = 0;                                         // tensor_dim2_stride lo
  g2[3] = 0;                                         // stride hi | tile_dim3=0
  i32x4 g3 = {0, 0, 0, 0};
  i32x8 gx = {0, 0, 0, 0, 0, 0, 0, 0};               // unused trailing group
  __builtin_amdgcn_tensor_load_to_lds(g0, g1, g2, g3, gx, /*cpol=*/0);
}

// ---------------------------------------------------------------------------
// SE(3) weight expansion: out[o][0..Cin-1] = W[o][..], out[o][Cin] = 1 - rowsum
// ---------------------------------------------------------------------------
__global__ void se3_expand_kernel(const float* __restrict__ W,
                                  float* __restrict__ out, int Cin) {
  __shared__ float red[256];
  const int o = blockIdx.x;
  const int t = threadIdx.x;
  float w = (t < Cin) ? W[o * Cin + t] : 0.0f;
  red[t] = w;
  __syncthreads();
#pragma unroll
  for (int s = 128; s > 0; s >>= 1) {
    if (t < s) red[t] += red[t + s];
    __syncthreads();
  }
  const float rowsum = red[0];
  if (t < Cin) out[o * (Cin + 1) + t] = w;
  if (t == Cin) out[o * (Cin + 1) + t] = 1.0f - rowsum;
}

// ---------------------------------------------------------------------------
// Fused main kernel
// ---------------------------------------------------------------------------
__global__ __launch_bounds__(256)
void fused_vn_kernel(const float* __restrict__ v_in, const float* __restrict__ s_in,
                     const float* __restrict__ Wv,   // se3(weight)      [128x128]
                     const float* __restrict__ Wc,   // se3(cross)       [128x128]
                     const float* __restrict__ Wf,   // se3(crossfc)     [128x256]
                     const float* __restrict__ svW, const float* __restrict__ svb,
                     const float* __restrict__ vsdir,
                     const float* __restrict__ vsW, const float* __restrict__ vsb,
                     const float* __restrict__ ssW, const float* __restrict__ ssb,
                     float* __restrict__ out_v, float* __restrict__ out_s) {
  constexpr int C = 128, Nn = 4096, NT = 16, LDV = 48;

  extern __shared__ float smem[];
  float* sV   = smem;              // [C][48]  V tile (later centered in place)
  float* sv1  = sV + C * LDV;      // [C][48]  v_out working buffer
  float* svd  = sv1 + C * LDV;     // [C][48]  v_dual / v_cross / dual_dir
  float* sS   = svd + C * LDV;     // [C][16]  S tile
  float* ss2v = sS + C * NT;       // [C][16]  s2v
  float* ssfv = ss2v + C * NT;     // [C][16]  scratch norms / s_from_v
  float* vmA  = ssfv + C * NT;     // [48]
  float* vmB  = vmA + 48;          // [48]
  float* colr = vmB + 48;          // [16]

  const int tid  = threadIdx.x;
  const int lane = tid & 31;
  const int wave = tid >> 5;       // 0..7 (wave32)
  const int m    = lane & 15;      // row (A) / column (B,C,D) within tile
  const int duo  = lane >> 4;      // half-wave select
  const int o0   = wave * 16;      // 16 output channels per wave
  const int b    = blockIdx.y;
  const int n0   = blockIdx.x * NT;

  // ---- stage V and S tiles into LDS via the Tensor Data Mover ------------
  // V tile: 3-D tile (n:16 contiguous, v:3 @ stride 4096, c:128 @ stride 12288)
  // lands contiguously in LDS = sV[c][v*16+n] (row stride 48 floats).
  // S tile: 2-D tile (n:16 contiguous, c:128 @ stride 4096) -> sS[c][n].
  if (wave == 0) {
    unsigned sV_off = (unsigned)(unsigned long long)(void*)sV;
    unsigned sS_off = (unsigned)(unsigned long long)(void*)sS;
    const float* vg = v_in + (size_t)b * C * 3 * Nn + n0;
    const float* sg = s_in + (size_t)b * C * Nn + n0;
    tdm_load_f32_tile(sV_off, vg, /*tile*/ NT, 3, C,
                      /*tdim*/ Nn, 3, C, /*strides*/ Nn, 3ull * Nn);
    tdm_load_f32_tile(sS_off, sg, /*tile*/ NT, C, 0,
                      /*tdim*/ Nn, C, 0, /*strides*/ Nn, 0);
  }
  // Warm L1/L2 with the weight rows this wave will stream as A-fragments
  // while the TDM transfer is in flight (global_prefetch_b8).
  __builtin_prefetch(Wv + m * C, 0, 3);
  __builtin_prefetch(Wc + m * C, 0, 3);
  __builtin_prefetch(svW + m * C, 0, 3);
  __builtin_prefetch(Wf + m * 2 * C, 0, 3);
  if (wave == 0) __builtin_amdgcn_s_wait_tensorcnt(0);
  __syncthreads();

  // ---- GEMM phase 1: vlin(Wv), vdual(Wc), s2v(svW)  (K=128) --------------
  {
    v8f av0 = {}, av1 = {}, av2 = {}, ad0 = {}, ad1 = {}, ad2 = {}, accS = {};
#pragma unroll 2
    for (int kk = 0; kk < C; kk += 4) {
      v2f aV = load_a(Wv, C, m, kk, duo);
      v2f aD = load_a(Wc, C, m, kk, duo);
      v2f aS = load_a(svW, C, m, kk, duo);
      v2f b0 = load_b(sV + 0,  LDV, m, kk, duo);
      v2f b1 = load_b(sV + 16, LDV, m, kk, duo);
      v2f b2 = load_b(sV + 32, LDV, m, kk, duo);
      v2f bS = load_b(sS, NT, m, kk, duo);
      av0 = wmma4(aV, b0, av0); av1 = wmma4(aV, b1, av1); av2 = wmma4(aV, b2, av2);
      ad0 = wmma4(aD, b0, ad0); ad1 = wmma4(aD, b1, ad1); ad2 = wmma4(aD, b2, ad2);
      accS = wmma4(aS, bS, accS);
    }
    store_tile(sv1 + 0,  LDV, o0, duo, m, av0);
    store_tile(sv1 + 16, LDV, o0, duo, m, av1);
    store_tile(sv1 + 32, LDV, o0, duo, m, av2);
    store_tile(svd + 0,  LDV, o0, duo, m, ad0);
    store_tile(svd + 16, LDV, o0, duo, m, ad1);
    store_tile(svd + 32, LDV, o0, duo, m, ad2);
#pragma unroll
    for (int r = 0; r < 8; ++r) {
      int row = o0 + r + 8 * duo;
      ss2v[row * NT + m] = accS[r] + svb[row];
    }
  }
  __syncthreads();

  // ---- column reductions: v_mean (pre), s2v channel-norm ------------------
  if (tid < 48) {
    float s = 0.0f;
    for (int c = 0; c < C; ++c) s += sv1[c * LDV + tid];
    vmA[tid] = s * (1.0f / C);
  } else if (tid >= 64 && tid < 80) {
    int n = tid - 64;
    float s = 0.0f;
    for (int c = 0; c < C; ++c) { float x = ss2v[c * NT + n]; s += x * x; }
    colr[n] = fmaxf(sqrtf(s), EPSN);
  }
  __syncthreads();

  // ---- v_out = (v - mean) * normalize(s2v) + mean ------------------------
  for (int idx = tid; idx < C * NT; idx += 256) {
    int c = idx >> 4, n = idx & 15;
    float sc = ss2v[idx] / colr[n];
#pragma unroll
    for (int v = 0; v < 3; ++v) {
      float mv = vmA[v * 16 + n];
      float x = sv1[c * LDV + v * 16 + n];
      sv1[c * LDV + v * 16 + n] = (x - mv) * sc + mv;
    }
  }
  __syncthreads();

  // ---- d_o = mean(v_dual), v_o = mean(v_out post-scale) ------------------
  if (tid < 48) {
    float s = 0.0f;
    for (int c = 0; c < C; ++c) s += svd[c * LDV + tid];
    vmA[tid] = s * (1.0f / C);
  } else if (tid >= 64 && tid < 112) {
    int j = tid - 64;
    float s = 0.0f;
    for (int c = 0; c < C; ++c) s += sv1[c * LDV + j];
    vmB[j] = s * (1.0f / C);
  }
  __syncthreads();

  // ---- channel_equi_vec_normalize pass 1: per-(c,n) vector norms ---------
  for (int idx = tid; idx < C * NT; idx += 256) {
    int c = idx >> 4, n = idx & 15;
    float ux = svd[c * LDV + n]       - vmA[n];
    float uy = svd[c * LDV + 16 + n]  - vmA[16 + n];
    float uz = svd[c * LDV + 32 + n]  - vmA[32 + n];
    ssfv[idx] = sqrtf(ux * ux + uy * uy + uz * uz);
  }
  __syncthreads();
  if (tid < 16) {
    float s = 0.0f;
    for (int c = 0; c < C; ++c) { float x = ssfv[c * NT + tid]; s += x * x; }
    colr[tid] = fmaxf(sqrtf(s), EPSN);
  }
  __syncthreads();

  // ---- cross product + add v_out; overwrite svd with v_cross -------------
  for (int idx = tid; idx < C * NT; idx += 256) {
    int c = idx >> 4, n = idx & 15;
    float un = ssfv[idx];
    float sc = (un / colr[n]) / fmaxf(un, EPSN);   // x_nn / ||dir||
    float ex = (svd[c * LDV + n]      - vmA[n])      * sc;
    float ey = (svd[c * LDV + 16 + n] - vmA[16 + n]) * sc;
    float ez = (svd[c * LDV + 32 + n] - vmA[32 + n]) * sc;
    float vx = sv1[c * LDV + n],      wx = vx - vmB[n];
    float vy = sv1[c * LDV + 16 + n], wy = vy - vmB[16 + n];
    float vz = sv1[c * LDV + 32 + n], wz = vz - vmB[32 + n];
    svd[c * LDV + n]      = (ey * wz - ez * wy) + vx;
    svd[c * LDV + 16 + n] = (ez * wx - ex * wz) + vy;
    svd[c * LDV + 32 + n] = (ex * wy - ey * wx) + vz;
  }
  __syncthreads();

  // ---- crossfc GEMM (K=256: rows 0-127 = v_cross, 128-255 = v_out) -------
  {
    v8f a0 = {}, a1 = {}, a2 = {};
#pragma unroll 2
    for (int kk = 0; kk < C; kk += 4) {
      v2f aW = load_a(Wf, 2 * C, m, kk, duo);
      v2f b0 = load_b(svd + 0,  LDV, m, kk, duo);
      v2f b1 = load_b(svd + 16, LDV, m, kk, duo);
      v2f b2 = load_b(svd + 32, LDV, m, kk, duo);
      a0 = wmma4(aW, b0, a0); a1 = wmma4(aW, b1, a1); a2 = wmma4(aW, b2, a2);
    }
#pragma unroll 2
    for (int kk = 0; kk < C; kk += 4) {
      v2f aW = load_a(Wf, 2 * C, m, kk + C, duo);
      v2f b0 = load_b(sv1 + 0,  LDV, m, kk, duo);
      v2f b1 = load_b(sv1 + 16, LDV, m, kk, duo);
      v2f b2 = load_b(sv1 + 32, LDV, m, kk, duo);
      a0 = wmma4(aW, b0, a0); a1 = wmma4(aW, b1, a1); a2 = wmma4(aW, b2, a2);
    }
#pragma unroll
    for (int r = 0; r < 8; ++r) {
      int row = o0 + r + 8 * duo;
      size_t base = ((size_t)(b * C + row) * 3) * Nn + n0 + m;
      out_v[base]          = a0[r];
      out_v[base + Nn]     = a1[r];
      out_v[base + 2 * Nn] = a2[r];
    }
  }

  // ---- scalar path: center V over channels -------------------------------
  if (tid < 48) {
    float s = 0.0f;
    for (int c = 0; c < C; ++c) s += sV[c * LDV + tid];
    vmA[tid] = s * (1.0f / C);
  }
  __syncthreads();
  for (int idx = tid; idx < C * NT; idx += 256) {
    int c = idx >> 4, n = idx & 15;
#pragma unroll
    for (int v = 0; v < 3; ++v) sV[c * LDV + v * 16 + n] -= vmA[v * 16 + n];
  }
  __syncthreads();

  // ---- dual_dir GEMM: vsdir @ v_sR  (writes svd) -------------------------
  {
    v8f d0 = {}, d1 = {}, d2 = {};
#pragma unroll 2
    for (int kk = 0; kk < C; kk += 4) {
      v2f aW = load_a(vsdir, C, m, kk, duo);
      v2f b0 = load_b(sV + 0,  LDV, m, kk, duo);
      v2f b1 = load_b(sV + 16, LDV, m, kk, duo);
      v2f b2 = load_b(sV + 32, LDV, m, kk, duo);
      d0 = wmma4(aW, b0, d0); d1 = wmma4(aW, b1, d1); d2 = wmma4(aW, b2, d2);
    }
    store_tile(svd + 0,  LDV, o0, duo, m, d0);
    store_tile(svd + 16, LDV, o0, duo, m, d1);
    store_tile(svd + 32, LDV, o0, duo, m, d2);
  }
  __syncthreads();

  // ---- s_from_v = normalize_ch( dot(v_sR, normalize3(dual_dir)) ) --------
  for (int idx = tid; idx < C * NT; idx += 256) {
    int c = idx >> 4, n = idx & 15;
    float dx = svd[c * LDV + n], dy = svd[c * LDV + 16 + n], dz = svd[c * LDV + 32 + n];
    float dn = sqrtf(dx * dx + dy * dy + dz * dz);
    float inv = 1.0f / fmaxf(dn, EPSN);
    float s = (sV[c * LDV + n] * dx + sV[c * LDV + 16 + n] * dy +
               sV[c * LDV + 32 + n] * dz) * inv;
    ssfv[idx] = s;
  }
  __syncthreads();
  if (tid < 16) {
    float s = 0.0f;
    for (int c = 0; c < C; ++c) { float x = ssfv[c * NT + tid]; s += x * x; }
    colr[tid] = fmaxf(sqrtf(s), EPSN);
  }
  __syncthreads();
  for (int idx = tid; idx < C * NT; idx += 256) ssfv[idx] /= colr[idx & 15];
  __syncthreads();

  // ---- final scalar GEMMs: vs_W @ s_from_v + ss_W @ s_in + biases --------
  {
    v8f aA = {}, aB = {};
#pragma unroll 2
    for (int kk = 0; kk < C; kk += 4) {
      v2f a1 = load_a(vsW, C, m, kk, duo);
      v2f b1 = load_b(ssfv, NT, m, kk, duo);
      v2f a2 = load_a(ssW, C, m, kk, duo);
      v2f b2 = load_b(sS, NT, m, kk, duo);
      aA = wmma4(a1, b1, aA);
      aB = wmma4(a2, b2, aB);
    }
#pragma unroll
    for (int r = 0; r < 8; ++r) {
      int row = o0 + r + 8 * duo;
      out_s[(size_t)(b * C + row) * Nn + n0 + m] = aA[r] + aB[r] + vsb[row] + ssb[row];
    }
  }
}

// ---------------------------------------------------------------------------
extern "C" void kernel_launch(void* const* d_in, const int* in_sizes, int n_in,
                              void* d_out, int out_size, void* d_ws, size_t ws_size,
                              hipStream_t stream) {
  (void)in_sizes; (void)n_in; (void)out_size; (void)ws_size;

  const float* v_input        = (const float*)d_in[0];
  const float* s_input        = (const float*)d_in[1];
  const float* weight         = (const float*)d_in[2];   // [128,127]
  const float* sv_W           = (const float*)d_in[3];   // [128,128]
  const float* sv_b           = (const float*)d_in[4];
  const float* cross_weight   = (const float*)d_in[5];   // [128,127]
  const float* crossfc_weight = (const float*)d_in[6];   // [128,255]
  const float* vsdir_weight   = (const float*)d_in[7];   // [128,128]
  const float* vs_W           = (const float*)d_in[8];
  const float* vs_b           = (const float*)d_in[9];
  const float* ss_W           = (const float*)d_in[10];
  const float* ss_b           = (const float*)d_in[11];

  float* ws = (float*)d_ws;
  float* Wv = ws;                  // 128*128
  float* Wc = ws + 128 * 128;      // 128*128
  float* Wf = ws + 2 * 128 * 128;  // 128*256

  se3_expand_kernel<<<128, 256, 0, stream>>>(weight, Wv, 127);
  se3_expand_kernel<<<128, 256, 0, stream>>>(cross_weight, Wc, 127);
  se3_expand_kernel<<<128, 256, 0, stream>>>(crossfc_weight, Wf, 255);

  float* out_v = (float*)d_out;
  float* out_s = out_v + (size_t)16 * 128 * 3 * 4096;

  const size_t smem_floats = 3 * (128 * 48) + 3 * (128 * 16) + 48 + 48 + 16;
  const size_t smem_bytes = smem_floats * sizeof(float);   // ~96.5 KB
  (void)hipFuncSetAttribute((const void*)fused_vn_kernel,
                            hipFuncAttributeMaxDynamicSharedMemorySize,
                            (int)smem_bytes);

  dim3 grid(4096 / 16, 16);
  fused_vn_kernel<<<grid, 256, smem_bytes, stream>>>(
      v_input, s_input, Wv, Wc, Wf, sv_W, sv_b, vsdir_weight,
      vs_W, vs_b, ss_W, ss_b, out_v, out_s);
}